// BilinearSentenceEncoder_25881472926223
// MI455X (gfx1250) — compile-verified
//
#include <hip/hip_runtime.h>
#include <math.h>

typedef float v2f __attribute__((ext_vector_type(2)));
typedef float v8f __attribute__((ext_vector_type(8)));

constexpr int Ls = 128;
constexpr int Bn = 4096;
constexpr int Dd = 300;
constexpr int TS = 302;        // padded LDS row stride (8B-aligned, good bank spread)

// ---------------------------------------------------------------------------
// Pass 1: per 16-row tile (all rows share one l), compute z = s^T W via
// f32 WMMA and reduce to three per-row dots: z.self, z.prev, z.next.
// ---------------------------------------------------------------------------
__global__ __launch_bounds__(128) void bil_scores_kernel(
    const float* __restrict__ S, const float* __restrict__ W,
    float* __restrict__ dotSelf, float* __restrict__ dotPrev,
    float* __restrict__ dotNext)
{
    __shared__ float sS[16 * TS];
    __shared__ float sP[16 * TS];
    __shared__ float sN[16 * TS];
    __shared__ float sDots[3 * 16];

    const int tid = threadIdx.x;
    const int t   = blockIdx.x;        // tile index over M = L*B rows
    const int l   = t >> 8;            // t / (B/16), B/16 = 256
    const int b0  = (t & 255) << 4;
    const size_t rowBase = (size_t)l * Bn + b0;

    // Stage self / prev / next 16x300 tiles into LDS (coalesced).
    for (int e = tid; e < 16 * Dd; e += 128) {
        int r = e / Dd, c = e - r * Dd;
        size_t g = (rowBase + r) * Dd + c;
        sS[r * TS + c] = S[g];
        sP[r * TS + c] = (l > 0)      ? S[g - (size_t)Bn * Dd] : 0.0f;
        sN[r * TS + c] = (l < Ls - 1) ? S[g + (size_t)Bn * Dd] : 0.0f;
    }
    if (tid < 48) sDots[tid] = 0.0f;
    __syncthreads();

    const int wave = tid >> 5;
    const int lane = tid & 31;
    const int half = lane >> 4;        // 0: rows 0..7 / K 0..1, 1: rows 8..15 / K 2..3
    const int l16  = lane & 15;

    float accS[8] = {0,0,0,0,0,0,0,0};
    float accP[8] = {0,0,0,0,0,0,0,0};
    float accN[8] = {0,0,0,0,0,0,0,0};

    // Waves split the 19 N-tiles (300 cols). For the last tile the cols >= 300
    // are CLAMPED to col 299 (in-bounds, branch-free loads); the resulting
    // garbage z-columns are masked to zero at the dot-fold stage instead.
    for (int nt = wave; nt < 19; nt += 4) {
        const int n0   = nt << 4;
        const int col  = n0 + l16;
        const bool colOK = (col < Dd);
        const int colC = colOK ? col : (Dd - 1);
        const float* wb = W + colC;    // column pointer into W[d,e] (row stride D)

        v8f acc0 = {}; v8f acc1 = {};  // two chains -> independent WMMAs

        int kt = 0;
        for (; kt < 74; kt += 2) {
            const int kA = (kt << 2) + (half << 1);
            v2f a0 = *(const v2f*)&sS[l16 * TS + kA];
            v2f a1 = *(const v2f*)&sS[l16 * TS + kA + 4];
            v2f b0v, b1v;
            b0v.x = wb[(size_t)(kA + 0) * Dd];
            b0v.y = wb[(size_t)(kA + 1) * Dd];
            b1v.x = wb[(size_t)(kA + 4) * Dd];
            b1v.y = wb[(size_t)(kA + 5) * Dd];
            acc0 = __builtin_amdgcn_wmma_f32_16x16x4_f32(
                false, a0, false, b0v, (short)0, acc0, false, false);
            acc1 = __builtin_amdgcn_wmma_f32_16x16x4_f32(
                false, a1, false, b1v, (short)0, acc1, false, false);
        }
        {   // final K step (kt == 74), K = 300 = 75*4 exactly
            const int kA = 296 + (half << 1);
            v2f a0 = *(const v2f*)&sS[l16 * TS + kA];
            v2f b0v;
            b0v.x = wb[(size_t)(kA + 0) * Dd];
            b0v.y = wb[(size_t)(kA + 1) * Dd];
            acc0 = __builtin_amdgcn_wmma_f32_16x16x4_f32(
                false, a0, false, b0v, (short)0, acc0, false, false);
        }

        // Fold z-tile into per-row dot partials. C/D layout: lane<16 -> row r,
        // lane>=16 -> row r+8, column = n0 + (lane&15). Out-of-range columns
        // contribute zero via the z mask (v_cndmask, no branch).
        #pragma unroll
        for (int r = 0; r < 8; ++r) {
            float z = acc0[r] + acc1[r];
            z = colOK ? z : 0.0f;
            int mloc = r + (half << 3);
            float vs = sS[mloc * TS + colC];
            float vp = sP[mloc * TS + colC];
            float vn = sN[mloc * TS + colC];
            accS[r] = fmaf(z, vs, accS[r]);
            accP[r] = fmaf(z, vp, accP[r]);
            accN[r] = fmaf(z, vn, accN[r]);
        }
    }

    // Reduce over the 16 lanes of each half-wave (halves hold disjoint rows).
    #pragma unroll
    for (int r = 0; r < 8; ++r) {
        #pragma unroll
        for (int off = 1; off < 16; off <<= 1) {
            accS[r] += __shfl_xor(accS[r], off, 32);
            accP[r] += __shfl_xor(accP[r], off, 32);
            accN[r] += __shfl_xor(accN[r], off, 32);
        }
    }
    if (l16 == 0) {
        #pragma unroll
        for (int r = 0; r < 8; ++r) {
            int mloc = r + (half << 3);
            atomicAdd(&sDots[0 * 16 + mloc], accS[r]);
            atomicAdd(&sDots[1 * 16 + mloc], accP[r]);
            atomicAdd(&sDots[2 * 16 + mloc], accN[r]);
        }
    }
    __syncthreads();

    if (tid < 16) {
        size_t m = rowBase + tid;
        dotSelf[m] = sDots[0 * 16 + tid];
        dotPrev[m] = sDots[1 * 16 + tid];
        dotNext[m] = sDots[2 * 16 + tid];
    }
}

// ---------------------------------------------------------------------------
// Pass 2: softmax over 3 channels + weighted neighbor combine.
// ---------------------------------------------------------------------------
__global__ __launch_bounds__(128) void combine_kernel(
    const float* __restrict__ S, const int* __restrict__ size,
    const float* __restrict__ dS, const float* __restrict__ dP,
    const float* __restrict__ dN, float* __restrict__ out)
{
    const size_t m = blockIdx.x;       // row index over L*B
    const int l = (int)(m >> 12);      // m / 4096
    const int b = (int)(m & 4095);
    const int sz = size[b];
    const float NEGINF = -__builtin_huge_valf();

    float x1 = dS[m];
    float symPrev = 0.0f, symNext = 0.0f;
    if (l > 0)      symPrev = 0.5f * (dP[m] + dN[m - Bn]);
    if (l < Ls - 1) symNext = 0.5f * (dN[m] + dP[m + Bn]);

    float x0 = (l >= 1 && l < sz)           ? symPrev : NEGINF;
    float x2 = (l <= Ls - 2 && l < sz - 1)  ? symNext : NEGINF;

    const float invD = 1.0f / (float)Dd;
    x0 *= invD; x1 *= invD; x2 *= invD;

    float mx = fmaxf(x1, fmaxf(x0, x2));   // x1 always finite
    float e0 = __expf(x0 - mx);
    float e1 = __expf(x1 - mx);
    float e2 = __expf(x2 - mx);
    float inv = 1.0f / (e0 + e1 + e2);
    float p0 = e0 * inv, p1 = e1 * inv, p2 = e2 * inv;

    const float* rs = S + m * (size_t)Dd;
    const float* rp = rs - (size_t)Bn * Dd;
    const float* rn = rs + (size_t)Bn * Dd;
    float* ro = out + m * (size_t)Dd;

    for (int d = threadIdx.x; d < Dd; d += 128) {
        float v = p1 * rs[d];
        if (l > 0)      v += p0 * rp[d];   // p0 == 0 exactly when l == 0
        if (l < Ls - 1) v += p2 * rn[d];   // p2 == 0 exactly when l == L-1
        ro[d] = v;
    }
}

// ---------------------------------------------------------------------------
extern "C" void kernel_launch(void* const* d_in, const int* in_sizes, int n_in,
                              void* d_out, int out_size, void* d_ws, size_t ws_size,
                              hipStream_t stream)
{
    const float* S    = (const float*)d_in[0];   // sentence [L,B,D] f32
    const int*   size = (const int*)  d_in[1];   // [B] int32
    const float* W    = (const float*)d_in[2];   // [D,D] f32
    float* out = (float*)d_out;

    const size_t M = (size_t)Ls * Bn;            // 524288 rows
    float* ws = (float*)d_ws;                    // needs 3*M*4 = 6.3 MB
    float* dS = ws;
    float* dP = ws + M;
    float* dN = ws + 2 * M;

    bil_scores_kernel<<<dim3((unsigned)(M / 16)), dim3(128), 0, stream>>>(
        S, W, dS, dP, dN);
    combine_kernel<<<dim3((unsigned)M), dim3(128), 0, stream>>>(
        S, size, dS, dP, dN, out);
}